// M3GNet_60919816127133
// MI455X (gfx1250) — compile-verified
//
#include <hip/hip_runtime.h>
#include <math.h>

#define DD 128          // feature dim
#define NBF 20          // angular basis
#define BLK 128         // 4 waves
#define ROWS 32         // rows (M) per block: 2 M-tiles per wave

typedef float v2f __attribute__((ext_vector_type(2)));
typedef float v8f __attribute__((ext_vector_type(8)));

__device__ __forceinline__ float sigm(float x) { return 1.0f / (1.0f + __expf(-x)); }
__device__ __forceinline__ float silu(float x) { return x * sigm(x); }
__device__ __forceinline__ float n2n(float x) {
    if (__builtin_isnan(x) || __builtin_isinf(x)) return 0.0f;
    return x;
}

// gfx1250 async memory->LDS copy, 16 bytes per lane, tracked by ASYNCcnt.
__device__ __forceinline__ void async_ld16(unsigned int lds_off, const void* gptr) {
    asm volatile("global_load_async_to_lds_b128 %0, %1, off"
                 :: "v"(lds_off), "v"((unsigned long long)(uintptr_t)gptr)
                 : "memory");
}
__device__ __forceinline__ void async_wait0() {
    asm volatile("s_wait_asynccnt 0x0" ::: "memory");
}
__device__ __forceinline__ unsigned int lds_off(const void* p) {
    return (unsigned int)(uintptr_t)p;   // low 32 bits of generic addr = LDS offset
}

// One wave computes a 32(M) x 32(N) output block (2 M-tiles x 2 N-tiles) of a
// [32 x K](LDS, stride lda) @ [K x N](global row-major, stride ldw) GEMM with
// V_WMMA_F32_16X16X4_F32.  B fragments are reused across both M-tiles.
__device__ __forceinline__ void wmma_2x2(const float* __restrict__ As, int lda,
                                         const float* __restrict__ W, int ldw,
                                         int K, int n0, int half, int l16,
                                         v8f& c00, v8f& c01, v8f& c10, v8f& c11) {
    for (int k0 = 0; k0 < K; k0 += 4) {
        const int kk = k0 + half * 2;   // A 16x4: lanes<16 -> K=k0,k0+1 ; lanes>=16 -> K=k0+2,k0+3
        v2f a0; a0.x = As[l16 * lda + kk];        a0.y = As[l16 * lda + kk + 1];
        v2f a1; a1.x = As[(16 + l16) * lda + kk]; a1.y = As[(16 + l16) * lda + kk + 1];
        v2f b0; b0.x = W[(size_t)kk * ldw + n0 + l16];
                b0.y = W[(size_t)(kk + 1) * ldw + n0 + l16];
        v2f b1; b1.x = W[(size_t)kk * ldw + n0 + 16 + l16];
                b1.y = W[(size_t)(kk + 1) * ldw + n0 + 16 + l16];
        c00 = __builtin_amdgcn_wmma_f32_16x16x4_f32(false, a0, false, b0, (short)0, c00, false, false);
        c01 = __builtin_amdgcn_wmma_f32_16x16x4_f32(false, a0, false, b1, (short)0, c01, false, false);
        c10 = __builtin_amdgcn_wmma_f32_16x16x4_f32(false, a1, false, b0, (short)0, c10, false, false);
        c11 = __builtin_amdgcn_wmma_f32_16x16x4_f32(false, a1, false, b1, (short)0, c11, false, false);
    }
}
#define ZERO8 {0,0,0,0,0,0,0,0}

// ---------------------------------------------------------------------------
// Stage 1: edge MLP + gate.
// ---------------------------------------------------------------------------
__global__ __launch_bounds__(BLK) void edge_mlp_kernel(
    const float* __restrict__ nodef, const float* __restrict__ edgef,
    const int* __restrict__ eidx, int E,
    const float* __restrict__ W1, const float* __restrict__ b1,
    const float* __restrict__ W2, const float* __restrict__ b2,
    const float* __restrict__ gW, const float* __restrict__ gb,
    float* __restrict__ edge1)
{
    __shared__ float As[ROWS][384];
    __shared__ float Hs[ROWS][DD];
    __shared__ int ssrc[ROWS], sdst[ROWS];

    const int tid  = threadIdx.x;
    const int base = blockIdx.x * ROWS;

    if (tid < ROWS) {
        int e = min(base + tid, E - 1);
        ssrc[tid] = eidx[e];
        sdst[tid] = eidx[E + e];
    }
    __syncthreads();

    for (int i = tid; i < ROWS * 96; i += BLK) {     // 96 float4 per row, async copy
        int row = i / 96, c4 = i % 96;
        int e = min(base + row, E - 1);
        const float* gp;
        if (c4 < 32)      gp = nodef + (size_t)ssrc[row] * DD + c4 * 4;
        else if (c4 < 64) gp = nodef + (size_t)sdst[row] * DD + (c4 - 32) * 4;
        else              gp = edgef + (size_t)e * DD + (c4 - 64) * 4;
        async_ld16(lds_off(&As[row][c4 * 4]), gp);
    }
    async_wait0();
    __syncthreads();

    const int lane = tid & 31, wv = tid >> 5;
    const int half = lane >> 4, l16 = lane & 15;
    const int n0 = wv * 32;

    v8f c00 = ZERO8, c01 = ZERO8, c10 = ZERO8, c11 = ZERO8;
    wmma_2x2(&As[0][0], 384, W1, DD, 384, n0, half, l16, c00, c01, c10, c11);

    const float bb0 = b1[n0 + l16], bb1 = b1[n0 + 16 + l16];
    #pragma unroll
    for (int r = 0; r < 8; ++r) {
        int r0 = half * 8 + r;
        Hs[r0][n0 + l16]           = silu(c00[r] + bb0);
        Hs[r0][n0 + 16 + l16]      = silu(c01[r] + bb1);
        Hs[16 + r0][n0 + l16]      = silu(c10[r] + bb0);
        Hs[16 + r0][n0 + 16 + l16] = silu(c11[r] + bb1);
    }
    __syncthreads();

    v8f d00 = ZERO8, d01 = ZERO8, d10 = ZERO8, d11 = ZERO8;
    wmma_2x2(&Hs[0][0], DD, W2, DD, DD, n0, half, l16, d00, d01, d10, d11);
    v8f g00 = ZERO8, g01 = ZERO8, g10 = ZERO8, g11 = ZERO8;
    wmma_2x2(&As[0][256], 384, gW, DD, DD, n0, half, l16, g00, g01, g10, g11);

    const int col0 = n0 + l16, col1 = n0 + 16 + l16;
    const float wb0 = b2[col0], wb1 = b2[col1];
    const float gb0 = gb[col0], gb1 = gb[col1];
    #pragma unroll
    for (int mt = 0; mt < 2; ++mt) {
        #pragma unroll
        for (int r = 0; r < 8; ++r) {
            int row = mt * 16 + half * 8 + r;
            int e = min(base + row, E - 1);
            float dv0 = mt ? d10[r] : d00[r], dv1 = mt ? d11[r] : d01[r];
            float gv0 = mt ? g10[r] : g00[r], gv1 = mt ? g11[r] : g01[r];
            float o0 = As[row][256 + col0] + (dv0 + wb0) * sigm(gv0 + gb0);
            float o1 = As[row][256 + col1] + (dv1 + wb1) * sigm(gv1 + gb1);
            edge1[(size_t)e * DD + col0] = o0;
            edge1[(size_t)e * DD + col1] = o1;
        }
    }
}

// ---------------------------------------------------------------------------
// Stage 2: three-body MLP, scatter into delta[e_ij]
// ---------------------------------------------------------------------------
__global__ __launch_bounds__(BLK) void threebody_kernel(
    const float* __restrict__ edge1, const float* __restrict__ evec,
    const int* __restrict__ tbe, int T, int E,
    const float* __restrict__ W1, const float* __restrict__ b1,
    const float* __restrict__ W2, const float* __restrict__ b2,
    float* __restrict__ delta)
{
    __shared__ float As[ROWS][280];                  // 256 edge feats + 20 angle (padded)
    __shared__ float Hs[ROWS][DD];
    __shared__ int sij[ROWS], sik[ROWS];

    const int tid  = threadIdx.x;
    const int base = blockIdx.x * ROWS;

    if (tid < ROWS) {
        int t = min(base + tid, T - 1);
        sij[tid] = tbe[(size_t)t * 2 + 0];
        sik[tid] = tbe[(size_t)t * 2 + 1];
    }
    __syncthreads();

    for (int i = tid; i < ROWS * 64; i += BLK) {     // 64 float4 per row, n2n applied
        int row = i >> 6, c4 = i & 63;
        const float* src = (c4 < 32) ? (edge1 + (size_t)sij[row] * DD + c4 * 4)
                                     : (edge1 + (size_t)sik[row] * DD + (c4 - 32) * 4);
        float4 v = *(const float4*)src;
        v.x = n2n(v.x); v.y = n2n(v.y); v.z = n2n(v.z); v.w = n2n(v.w);
        *(float4*)&As[row][c4 * 4] = v;
    }
    if (tid < ROWS) {                                 // angle features
        int row = tid;
        const float* vij = evec + (size_t)sij[row] * 3;
        const float* vik = evec + (size_t)sik[row] * 3;
        float ax = vij[0], ay = vij[1], az = vij[2];  // negation cancels
        float bx = vik[0], by = vik[1], bz = vik[2];
        float lij = fmaxf(sqrtf(ax*ax + ay*ay + az*az), 1e-6f);
        float lik = fmaxf(sqrtf(bx*bx + by*by + bz*bz), 1e-6f);
        float c = (ax*bx + ay*by + az*bz) / (lij * lik);
        c = fminf(fmaxf(c, -1.0f), 1.0f);
        float p[5];
        p[0] = 1.0f; p[1] = c; p[2] = 0.5f * (3.0f*c*c - 1.0f);
        p[3] = 0.5f * (5.0f*c*c*c - 3.0f*c);
        p[4] = 0.125f * (35.0f*c*c*c*c - 30.0f*c*c + 3.0f);
        const float PI = 3.14159265358979323846f;
        float rad[4];
        #pragma unroll
        for (int n = 0; n < 4; ++n) rad[n] = sinf((float)(n + 1) * PI * lij / 5.0f) / lij;
        #pragma unroll
        for (int a = 0; a < 5; ++a)
            #pragma unroll
            for (int n = 0; n < 4; ++n)
                As[row][256 + a * 4 + n] = n2n(p[a] * rad[n]);
    }
    __syncthreads();

    const int lane = tid & 31, wv = tid >> 5;
    const int half = lane >> 4, l16 = lane & 15;
    const int n0 = wv * 32;

    v8f c00 = ZERO8, c01 = ZERO8, c10 = ZERO8, c11 = ZERO8;
    wmma_2x2(&As[0][0], 280, W1, DD, 276, n0, half, l16, c00, c01, c10, c11);

    const float bb0 = b1[n0 + l16], bb1 = b1[n0 + 16 + l16];
    #pragma unroll
    for (int r = 0; r < 8; ++r) {
        int r0 = half * 8 + r;
        Hs[r0][n0 + l16]           = silu(c00[r] + bb0);
        Hs[r0][n0 + 16 + l16]      = silu(c01[r] + bb1);
        Hs[16 + r0][n0 + l16]      = silu(c10[r] + bb0);
        Hs[16 + r0][n0 + 16 + l16] = silu(c11[r] + bb1);
    }
    __syncthreads();

    v8f d00 = ZERO8, d01 = ZERO8, d10 = ZERO8, d11 = ZERO8;
    wmma_2x2(&Hs[0][0], DD, W2, DD, DD, n0, half, l16, d00, d01, d10, d11);

    const int col0 = n0 + l16, col1 = n0 + 16 + l16;
    const float wb0 = b2[col0], wb1 = b2[col1];
    #pragma unroll
    for (int mt = 0; mt < 2; ++mt) {
        #pragma unroll
        for (int r = 0; r < 8; ++r) {
            int row = mt * 16 + half * 8 + r;
            if (base + row < T) {
                size_t eo = (size_t)sij[row] * DD;
                float dv0 = mt ? d10[r] : d00[r], dv1 = mt ? d11[r] : d01[r];
                atomicAdd(&delta[eo + col0], dv0 + wb0);
                atomicAdd(&delta[eo + col1], dv1 + wb1);
            }
        }
    }
}

// ---------------------------------------------------------------------------
// Stage 3: edge finalize + message accumulation
// ---------------------------------------------------------------------------
__global__ __launch_bounds__(BLK) void edge_final_kernel(
    const float* __restrict__ delta,
    const float* __restrict__ updW, const float* __restrict__ updb,
    const float* __restrict__ tbgW, const float* __restrict__ tbgb,
    const int* __restrict__ eidx, int E,
    float* __restrict__ edge_io,        // edge1 in, final edge out (in place)
    float* __restrict__ msg, float* __restrict__ deg)
{
    __shared__ float Ad[ROWS][DD];
    __shared__ float Ae[ROWS][DD];
    __shared__ int sdst[ROWS];

    const int tid  = threadIdx.x;
    const int base = blockIdx.x * ROWS;

    if (tid < ROWS) {
        int e = min(base + tid, E - 1);
        sdst[tid] = eidx[E + e];
        if (base + tid < E) atomicAdd(&deg[sdst[tid]], 1.0f);
    }
    for (int i = tid; i < ROWS * 32; i += BLK) {     // async copy, contiguous rows
        int row = i >> 5, c4 = i & 31;
        int e = min(base + row, E - 1);
        async_ld16(lds_off(&Ad[row][c4 * 4]), delta   + (size_t)e * DD + c4 * 4);
        async_ld16(lds_off(&Ae[row][c4 * 4]), edge_io + (size_t)e * DD + c4 * 4);
    }
    async_wait0();
    __syncthreads();

    const int lane = tid & 31, wv = tid >> 5;
    const int half = lane >> 4, l16 = lane & 15;
    const int n0 = wv * 32;

    v8f u00 = ZERO8, u01 = ZERO8, u10 = ZERO8, u11 = ZERO8;
    wmma_2x2(&Ad[0][0], DD, updW, DD, DD, n0, half, l16, u00, u01, u10, u11);
    v8f g00 = ZERO8, g01 = ZERO8, g10 = ZERO8, g11 = ZERO8;
    wmma_2x2(&Ae[0][0], DD, tbgW, DD, DD, n0, half, l16, g00, g01, g10, g11);

    const int col0 = n0 + l16, col1 = n0 + 16 + l16;
    const float ub0 = updb[col0], ub1 = updb[col1];
    const float tb0 = tbgb[col0], tb1 = tbgb[col1];
    #pragma unroll
    for (int mt = 0; mt < 2; ++mt) {
        #pragma unroll
        for (int r = 0; r < 8; ++r) {
            int row = mt * 16 + half * 8 + r;
            int e = min(base + row, E - 1);
            float uv0 = mt ? u10[r] : u00[r], uv1 = mt ? u11[r] : u01[r];
            float gv0 = mt ? g10[r] : g00[r], gv1 = mt ? g11[r] : g01[r];
            float o0 = Ae[row][col0] + n2n(uv0 + ub0) * sigm(gv0 + tb0);
            float o1 = Ae[row][col1] + n2n(uv1 + ub1) * sigm(gv1 + tb1);
            edge_io[(size_t)e * DD + col0] = o0;
            edge_io[(size_t)e * DD + col1] = o1;
            if (base + row < E) {
                size_t mo = (size_t)sdst[row] * DD;
                atomicAdd(&msg[mo + col0], o0);
                atomicAdd(&msg[mo + col1], o1);
            }
        }
    }
}

// ---------------------------------------------------------------------------
// Stage 4: node MLP
// ---------------------------------------------------------------------------
__global__ __launch_bounds__(BLK) void node_mlp_kernel(
    const float* __restrict__ nodef, const float* __restrict__ msg,
    const float* __restrict__ deg, int N,
    const float* __restrict__ W1, const float* __restrict__ b1,
    const float* __restrict__ W2, const float* __restrict__ b2,
    float* __restrict__ node_out)
{
    __shared__ float As[ROWS][256];
    __shared__ float Hs[ROWS][DD];
    __shared__ float sdeg[ROWS];

    const int tid  = threadIdx.x;
    const int base = blockIdx.x * ROWS;

    if (tid < ROWS) {
        int n = min(base + tid, N - 1);
        sdeg[tid] = fmaxf(deg[n], 1.0f);
    }
    for (int i = tid; i < ROWS * 32; i += BLK) {     // node feats: pure copy, async
        int row = i >> 5, c4 = i & 31;
        int n = min(base + row, N - 1);
        async_ld16(lds_off(&As[row][c4 * 4]), nodef + (size_t)n * DD + c4 * 4);
    }
    __syncthreads();                                  // sdeg visible
    for (int i = tid; i < ROWS * 32; i += BLK) {     // msg: normalize by degree
        int row = i >> 5, c4 = i & 31;
        int n = min(base + row, N - 1);
        float4 v = ((const float4*)(msg + (size_t)n * DD))[c4];
        float inv = 1.0f / sdeg[row];
        v.x *= inv; v.y *= inv; v.z *= inv; v.w *= inv;
        *(float4*)&As[row][128 + c4 * 4] = v;
    }
    async_wait0();
    __syncthreads();

    const int lane = tid & 31, wv = tid >> 5;
    const int half = lane >> 4, l16 = lane & 15;
    const int n0 = wv * 32;

    v8f c00 = ZERO8, c01 = ZERO8, c10 = ZERO8, c11 = ZERO8;
    wmma_2x2(&As[0][0], 256, W1, DD, 256, n0, half, l16, c00, c01, c10, c11);

    const float bb0 = b1[n0 + l16], bb1 = b1[n0 + 16 + l16];
    #pragma unroll
    for (int r = 0; r < 8; ++r) {
        int r0 = half * 8 + r;
        Hs[r0][n0 + l16]           = silu(c00[r] + bb0);
        Hs[r0][n0 + 16 + l16]      = silu(c01[r] + bb1);
        Hs[16 + r0][n0 + l16]      = silu(c10[r] + bb0);
        Hs[16 + r0][n0 + 16 + l16] = silu(c11[r] + bb1);
    }
    __syncthreads();

    v8f d00 = ZERO8, d01 = ZERO8, d10 = ZERO8, d11 = ZERO8;
    wmma_2x2(&Hs[0][0], DD, W2, DD, DD, n0, half, l16, d00, d01, d10, d11);

    const int col0 = n0 + l16, col1 = n0 + 16 + l16;
    const float wb0 = b2[col0], wb1 = b2[col1];
    #pragma unroll
    for (int mt = 0; mt < 2; ++mt) {
        #pragma unroll
        for (int r = 0; r < 8; ++r) {
            int row = mt * 16 + half * 8 + r;
            int n = min(base + row, N - 1);
            float dv0 = mt ? d10[r] : d00[r], dv1 = mt ? d11[r] : d01[r];
            node_out[(size_t)n * DD + col0] = As[row][col0] + dv0 + wb0;
            node_out[(size_t)n * DD + col1] = As[row][col1] + dv1 + wb1;
        }
    }
}

// ---------------------------------------------------------------------------
extern "C" void kernel_launch(void* const* d_in, const int* in_sizes, int n_in,
                              void* d_out, int out_size, void* d_ws, size_t ws_size,
                              hipStream_t stream) {
    const float* node_feats = (const float*)d_in[0];
    const float* edge_feats = (const float*)d_in[1];
    const float* edge_vecs  = (const float*)d_in[2];
    const int*   edge_index = (const int*)d_in[3];
    const int*   tb_edge    = (const int*)d_in[5];
    const float* edge_W1 = (const float*)d_in[6];
    const float* edge_b1 = (const float*)d_in[7];
    const float* edge_W2 = (const float*)d_in[8];
    const float* edge_b2 = (const float*)d_in[9];
    const float* egate_W = (const float*)d_in[10];
    const float* egate_b = (const float*)d_in[11];
    const float* phi_W1  = (const float*)d_in[12];
    const float* phi_b1  = (const float*)d_in[13];
    const float* phi_W2  = (const float*)d_in[14];
    const float* phi_b2  = (const float*)d_in[15];
    const float* upd_W   = (const float*)d_in[16];
    const float* upd_b   = (const float*)d_in[17];
    const float* tbg_W   = (const float*)d_in[18];
    const float* tbg_b   = (const float*)d_in[19];
    const float* node_W1 = (const float*)d_in[20];
    const float* node_b1 = (const float*)d_in[21];
    const float* node_W2 = (const float*)d_in[22];
    const float* node_b2 = (const float*)d_in[23];

    const int N = in_sizes[0] / DD;
    const int E = in_sizes[1] / DD;
    const int T = in_sizes[5] / 2;

    float* out_node = (float*)d_out;                       // (N, D)
    float* out_edge = (float*)d_out + (size_t)N * DD;      // (E, D) — also edge1 staging

    char*  ws    = (char*)d_ws;
    float* delta = (float*)ws;                                             // E*D
    float* msg   = (float*)(ws + (size_t)E * DD * sizeof(float));          // N*D
    float* deg   = (float*)(ws + ((size_t)E + N) * DD * sizeof(float));    // N

    hipMemsetAsync(delta, 0, (size_t)E * DD * sizeof(float), stream);
    hipMemsetAsync(msg,   0, (size_t)N * DD * sizeof(float), stream);
    hipMemsetAsync(deg,   0, (size_t)N * sizeof(float), stream);

    dim3 blk(BLK);
    edge_mlp_kernel<<<(E + ROWS - 1) / ROWS, blk, 0, stream>>>(
        node_feats, edge_feats, edge_index, E,
        edge_W1, edge_b1, edge_W2, edge_b2, egate_W, egate_b, out_edge);

    threebody_kernel<<<(T + ROWS - 1) / ROWS, blk, 0, stream>>>(
        out_edge, edge_vecs, tb_edge, T, E,
        phi_W1, phi_b1, phi_W2, phi_b2, delta);

    edge_final_kernel<<<(E + ROWS - 1) / ROWS, blk, 0, stream>>>(
        delta, upd_W, upd_b, tbg_W, tbg_b, edge_index, E,
        out_edge, msg, deg);

    node_mlp_kernel<<<(N + ROWS - 1) / ROWS, blk, 0, stream>>>(
        node_feats, msg, deg, N,
        node_W1, node_b1, node_W2, node_b2, out_node);
}